// MLPcache_19043884990814
// MI455X (gfx1250) — compile-verified
//
#include <hip/hip_runtime.h>
#include <cstdint>

// Dims fixed by the reference.
#define NRUN    64
#define KSAL    64
#define HIDDEN  4096
#define SLOTS   128
#define ROWS    512
#define THREADS 256

// ---------------------------------------------------------------------------
// Kernel 1: reset the winner map (SLOTS*ROWS ints) to -1. Must run every call
// (graph replay re-executes all three kernels; no state may persist in d_ws).
// ---------------------------------------------------------------------------
__global__ void win_init_kernel(int* __restrict__ win) {
    win[(size_t)blockIdx.x * THREADS + threadIdx.x] = -1;
}

// ---------------------------------------------------------------------------
// Kernel 2: scatter the update indices. Last-write-wins for duplicate row
// indices == max flattened update index (i*KSAL + j), computed via atomicMax.
// Winner map is indexed by cache *slot* so it stays correct for any
// running_seqs content.
// ---------------------------------------------------------------------------
__global__ void win_scatter_kernel(const int* __restrict__ run,
                                   const int* __restrict__ idx,
                                   int* __restrict__ win) {
    const int t = blockIdx.x * THREADS + threadIdx.x;  // t = i*KSAL + j
    const int i = t >> 6;                              // KSAL == 64
    const int s = run[i];
    const int r = idx[t];
    atomicMax(&win[s * ROWS + r], t);
}

// ---------------------------------------------------------------------------
// Kernel 3: stream one 16KB row per block. Source = winning x row (scattered)
// or the original cache row. Data path is the CDNA5 async DMA pipe:
//   global_load_async_to_lds_b128  ->  s_wait_asynccnt 0
//   global_store_async_from_lds_b128   (drained by s_endpgm's implicit wait)
// Per lane: 4 x 16B in, 4 x 16B out; per wave 4 x 512B transfers in flight.
// NT hints: 1GB streamed >> 192MB L2, avoid cache thrash.
// ---------------------------------------------------------------------------
__global__ __launch_bounds__(THREADS)
void row_copy_kernel(const float* __restrict__ x,
                     const float* __restrict__ cache,
                     const int* __restrict__ run,
                     const int* __restrict__ win,
                     float* __restrict__ out)
{
    __shared__ float lbuf[HIDDEN];                    // 16 KB staging

    const int row = blockIdx.x;                       // 0 .. NRUN*ROWS-1
    const int i   = row >> 9;                         // ROWS == 512
    const int r   = row & (ROWS - 1);
    const int s   = run[i];
    const int w   = win[s * ROWS + r];

    const float* src = (w >= 0)
        ? (x + (size_t)w * HIDDEN)                    // w = i*KSAL + j -> x row
        : (cache + ((size_t)s * ROWS + r) * HIDDEN);  // untouched cache row
    float* dst = out + (size_t)row * HIDDEN;

    const int t = threadIdx.x;
    const uint64_t g_src = (uint64_t)(uintptr_t)src + (uint64_t)t * 16u;
    const uint64_t g_dst = (uint64_t)(uintptr_t)dst + (uint64_t)t * 16u;
    // Low 32 bits of the flat shared-memory address == LDS byte offset
    // (flat LDS aperture).
    const uint32_t l_off = (uint32_t)(uintptr_t)(&lbuf[0]) + (uint32_t)t * 16u;

#pragma unroll
    for (int k = 0; k < 4; ++k) {
        asm volatile("global_load_async_to_lds_b128 %0, %1, off th:TH_LOAD_NT"
                     :: "v"(l_off + (uint32_t)(k * 4096)),
                        "v"(g_src + (uint64_t)(k * 4096))
                     : "memory");
    }
    // Ensure this wave's LDS staging writes have landed before reading back.
    asm volatile("s_wait_asynccnt 0" ::: "memory");
#pragma unroll
    for (int k = 0; k < 4; ++k) {
        asm volatile("global_store_async_from_lds_b128 %0, %1, off th:TH_STORE_NT"
                     :: "v"(g_dst + (uint64_t)(k * 4096)),
                        "v"(l_off + (uint32_t)(k * 4096))
                     : "memory");
    }
    // s_endpgm performs an implicit wait-idle: async stores drain before the
    // workgroup's LDS is deallocated.
}

// ---------------------------------------------------------------------------
extern "C" void kernel_launch(void* const* d_in, const int* in_sizes, int n_in,
                              void* d_out, int out_size, void* d_ws, size_t ws_size,
                              hipStream_t stream) {
    const float* x     = (const float*)d_in[0];  // [64,64,4096] f32
    const float* cache = (const float*)d_in[1];  // [128,512,4096] f32
    const int*   run   = (const int*)d_in[2];    // [64] i32
    const int*   idx   = (const int*)d_in[3];    // [64,64] i32
    float*       out   = (float*)d_out;          // [64,512,4096] f32
    int*         win   = (int*)d_ws;             // 128*512 ints = 256 KB

    (void)in_sizes; (void)n_in; (void)out_size; (void)ws_size;

    win_init_kernel<<<(SLOTS * ROWS) / THREADS, THREADS, 0, stream>>>(win);
    win_scatter_kernel<<<(NRUN * KSAL) / THREADS, THREADS, 0, stream>>>(run, idx, win);
    row_copy_kernel<<<NRUN * ROWS, THREADS, 0, stream>>>(x, cache, run, win, out);
}